// CachedAttention_19765439496474
// MI455X (gfx1250) — compile-verified
//
#include <hip/hip_runtime.h>
#include <math.h>

#define DMODEL 2048
#define NHEADS 16
#define HDIM   128
#define BSZ    16
#define SEQ    4096
#define SPLIT  8
#define CHUNK  (SEQ / SPLIT)

typedef float v2f __attribute__((ext_vector_type(2)));
typedef float v4f __attribute__((ext_vector_type(4)));
typedef float v8f __attribute__((ext_vector_type(8)));

// ---------------------------------------------------------------------------
// Y(16 x 2048) = X(16 x 2048) @ W(2048 x 2048)^T   (torch Linear, no bias)
// One block = 4 waves; block computes one 16-column tile of Y.
// Each wave accumulates a K-chunk of 512 with V_WMMA_F32_16X16X4_F32, then a
// deterministic LDS tree-sum combines the 4 partial accumulators.
//
// A (16x4, MxK) per-lane: lanes 0-15 -> M=lane, K={k,k+1}; lanes 16-31 -> K={k+2,k+3}
// B (4x16, KxN) per-lane: lanes 0-15 -> N=lane, K={k,k+1}; lanes 16-31 -> K={k+2,k+3}
// Both reduce to one contiguous 8-byte load per lane per step.
// The wave id is broadcast via readfirstlane so the K-loop bound is scalar:
// the backend emits a uniform s_cmp/s_cbranch loop and EXEC stays all-1s
// around the WMMAs (ISA requirement), with no per-iteration v_cmp/EXEC ops.
// ---------------------------------------------------------------------------
__global__ __launch_bounds__(128) void gemm16_xWt(const float* __restrict__ X,
                                                  const float* __restrict__ W,
                                                  float* __restrict__ Y) {
  const int lane = threadIdx.x & 31;
  const int wave = __builtin_amdgcn_readfirstlane(threadIdx.x >> 5);  // scalar
  const int row  = lane & 15;   // M index for A, N index for B/D
  const int hi   = lane >> 4;   // half-wave: K offset 0 or 2
  const int nbase = blockIdx.x * 16;

  const float* __restrict__ xrow = X + (size_t)row * DMODEL;            // A row M=row
  const float* __restrict__ wrow = W + (size_t)(nbase + row) * DMODEL;  // B col N=row

  v8f c = {0.f, 0.f, 0.f, 0.f, 0.f, 0.f, 0.f, 0.f};
  const int k0 = wave * (DMODEL / 4);
  const int k1 = k0 + (DMODEL / 4);
#pragma unroll 8
  for (int k = k0; k < k1; k += 4) {
    v2f a = *(const v2f*)(xrow + k + 2 * hi);
    v2f b = *(const v2f*)(wrow + k + 2 * hi);
    // (neg_a, A, neg_b, B, c_mod, C, reuse_a, reuse_b)
    c = __builtin_amdgcn_wmma_f32_16x16x4_f32(false, a, false, b, (short)0, c,
                                              false, false);
  }

  __shared__ float red[4][256];
#pragma unroll
  for (int r = 0; r < 8; ++r) red[wave][r * 32 + lane] = c[r];
  __syncthreads();

  if (wave == 0) {
#pragma unroll
    for (int r = 0; r < 8; ++r) {
      float s = red[0][r * 32 + lane] + red[1][r * 32 + lane] +
                red[2][r * 32 + lane] + red[3][r * 32 + lane];
      // D layout: VGPR r holds (M = r + 8*hi, N = row)
      Y[(size_t)(r + 8 * hi) * DMODEL + nbase + row] = s;
    }
  }
}

// ---------------------------------------------------------------------------
// Flash-decode partial pass: one block per (batch, head, seq-split).
// 8 waves stream 512 K/V rows; each wave-load is exactly one 512-byte row
// (lane l holds elements 4l..4l+3) -> perfectly coalesced b128 NT loads
// (cache is 1 GB >> 192 MB L2, so non-temporal avoids thrashing).
// Online softmax kept per-wave; 8 per-wave partials merged in LDS into one
// (m, l, acc[128]) partial per block, written to workspace.
// ---------------------------------------------------------------------------
__global__ __launch_bounds__(256) void attn_partial(
    const float* __restrict__ cache_k, const float* __restrict__ cache_v,
    const float* __restrict__ qbuf, float* __restrict__ pm,
    float* __restrict__ pl, float* __restrict__ pacc) {
  const int split = blockIdx.x;
  const int h = blockIdx.y;
  const int b = blockIdx.z;
  const int bh = b * NHEADS + h;
  const int lane = threadIdx.x & 31;
  const int wave = __builtin_amdgcn_readfirstlane(threadIdx.x >> 5);  // scalar
  const float scale = 0.08838834764831845f;  // 128^-0.5

  const float* qh = qbuf + (size_t)b * DMODEL + h * HDIM;
  const v4f q = *(const v4f*)(qh + lane * 4);

  const float* kb = cache_k + (size_t)bh * SEQ * HDIM;
  const float* vb = cache_v + (size_t)bh * SEQ * HDIM;

  float m = -3.0e38f, l = 0.f;
  v4f acc = {0.f, 0.f, 0.f, 0.f};

  const int s0 = split * CHUNK;
  for (int s = s0 + wave; s < s0 + CHUNK; s += 8) {
    const float* krow = kb + (size_t)s * HDIM;
    const float* vrow = vb + (size_t)s * HDIM;
    __builtin_prefetch(krow + 8 * HDIM + lane * 4, 0, 1);  // global_prefetch_b8
    __builtin_prefetch(vrow + 8 * HDIM + lane * 4, 0, 1);

    v4f kv = __builtin_nontemporal_load((const v4f*)krow + lane);
    float part = q.x * kv.x + q.y * kv.y + q.z * kv.z + q.w * kv.w;
#pragma unroll
    for (int off = 16; off >= 1; off >>= 1) part += __shfl_xor(part, off, 32);

    const float score = part * scale;
    const float mn = fmaxf(m, score);
    const float corr = __expf(m - mn);
    const float p = __expf(score - mn);
    v4f vv = __builtin_nontemporal_load((const v4f*)vrow + lane);
    l = l * corr + p;
    acc.x = acc.x * corr + p * vv.x;
    acc.y = acc.y * corr + p * vv.y;
    acc.z = acc.z * corr + p * vv.z;
    acc.w = acc.w * corr + p * vv.w;
    m = mn;
  }

  __shared__ float sm[8], sl[8], sacc[8][HDIM];
  sacc[wave][lane * 4 + 0] = acc.x;
  sacc[wave][lane * 4 + 1] = acc.y;
  sacc[wave][lane * 4 + 2] = acc.z;
  sacc[wave][lane * 4 + 3] = acc.w;
  if (lane == 0) { sm[wave] = m; sl[wave] = l; }
  __syncthreads();

  const int t = threadIdx.x;
  if (t < HDIM) {
    float M = sm[0];
#pragma unroll
    for (int w = 1; w < 8; ++w) M = fmaxf(M, sm[w]);
    float L = 0.f, a = 0.f;
#pragma unroll
    for (int w = 0; w < 8; ++w) {
      const float e = __expf(sm[w] - M);
      L += e * sl[w];
      a += e * sacc[w][t];
    }
    const size_t pidx = (size_t)bh * SPLIT + split;
    if (t == 0) { pm[pidx] = M; pl[pidx] = L; }
    pacc[pidx * HDIM + t] = a;
  }
}

// ---------------------------------------------------------------------------
// Merge SPLIT partials + the newly projected token (a one-row partial with
// m = score_new, l = 1, acc = v_new), normalize, write per-head output.
// One block (128 threads = one component each) per (batch, head).
// ---------------------------------------------------------------------------
__global__ __launch_bounds__(128) void attn_reduce(
    const float* __restrict__ qbuf, const float* __restrict__ kbuf,
    const float* __restrict__ vbuf, const float* __restrict__ pm,
    const float* __restrict__ pl, const float* __restrict__ pacc,
    float* __restrict__ attn_out) {
  const int bh = blockIdx.x;
  const int b = bh / NHEADS;
  const int h = bh % NHEADS;
  const int t = threadIdx.x;
  const float scale = 0.08838834764831845f;

  const float* qh = qbuf + (size_t)b * DMODEL + h * HDIM;
  const float* kh = kbuf + (size_t)b * DMODEL + h * HDIM;
  const float* vh = vbuf + (size_t)b * DMODEL + h * HDIM;

  __shared__ float sred[HDIM];
  sred[t] = qh[t] * kh[t];
  __syncthreads();
  for (int off = HDIM / 2; off >= 1; off >>= 1) {
    if (t < off) sred[t] += sred[t + off];
    __syncthreads();
  }
  const float s_new = sred[0] * scale;

  const float* m8 = pm + (size_t)bh * SPLIT;
  const float* l8 = pl + (size_t)bh * SPLIT;
  float M = s_new;
#pragma unroll
  for (int w = 0; w < SPLIT; ++w) M = fmaxf(M, m8[w]);
  const float en = __expf(s_new - M);
  float L = en;            // new token: l = 1
  float a = en * vh[t];    // new token: acc = v_new
#pragma unroll
  for (int w = 0; w < SPLIT; ++w) {
    const float e = __expf(m8[w] - M);
    L += e * l8[w];
    a += e * pacc[((size_t)bh * SPLIT + w) * HDIM + t];
  }
  attn_out[(size_t)b * DMODEL + h * HDIM + t] = a / L;
}

// ---------------------------------------------------------------------------
extern "C" void kernel_launch(void* const* d_in, const int* in_sizes, int n_in,
                              void* d_out, int out_size, void* d_ws,
                              size_t ws_size, hipStream_t stream) {
  (void)in_sizes; (void)n_in; (void)out_size; (void)ws_size;
  const float* x       = (const float*)d_in[0];
  const float* cache_k = (const float*)d_in[1];
  const float* cache_v = (const float*)d_in[2];
  const float* W_q     = (const float*)d_in[3];
  const float* W_k     = (const float*)d_in[4];
  const float* W_v     = (const float*)d_in[5];
  const float* W_o     = (const float*)d_in[6];
  float* out = (float*)d_out;

  float* ws = (float*)d_ws;
  float* qbuf     = ws;                       // 16*2048
  float* kbuf     = ws + 32768;               // 16*2048
  float* vbuf     = ws + 65536;               // 16*2048
  float* attn_out = ws + 98304;               // 16*2048
  float* pm       = ws + 131072;              // 256*8
  float* pl       = ws + 133120;              // 256*8
  float* pacc     = ws + 135168;              // 256*8*128

  // q/k/v projections (WMMA f32)
  gemm16_xWt<<<dim3(DMODEL / 16), 128, 0, stream>>>(x, W_q, qbuf);
  gemm16_xWt<<<dim3(DMODEL / 16), 128, 0, stream>>>(x, W_k, kbuf);
  gemm16_xWt<<<dim3(DMODEL / 16), 128, 0, stream>>>(x, W_v, vbuf);

  // HBM-streaming flash-decode over the 1.07 GB KV cache
  attn_partial<<<dim3(SPLIT, NHEADS, BSZ), 256, 0, stream>>>(
      cache_k, cache_v, qbuf, pm, pl, pacc);
  attn_reduce<<<dim3(BSZ * NHEADS), HDIM, 0, stream>>>(
      qbuf, kbuf, vbuf, pm, pl, pacc, attn_out);

  // output projection (WMMA f32) -> d_out
  gemm16_xWt<<<dim3(DMODEL / 16), 128, 0, stream>>>(attn_out, W_o, out);
}